// DirectedGNN_87548613362348
// MI455X (gfx1250) — compile-verified
//
#include <hip/hip_runtime.h>

// ---------------------------------------------------------------------------
// Directed 2-layer GraphSAGE for MI455X (gfx1250, wave32).
//   edge phase : native global_atomic_add_f32 scatter (1 wave per edge)
//   node phase : LDS-staged 16-row tiles + V_WMMA_F32_16X16X4_F32 dual GEMM
// ---------------------------------------------------------------------------

typedef __attribute__((ext_vector_type(2))) float v2f;
typedef __attribute__((ext_vector_type(8))) float v8f;

#define DIM 128
#define LPITCH 132   // 128 + 4 pad: stride-132 rows -> conflict-free column reads

// ---------------------------------------------------------------- zero fill
__global__ void gnn_zero_f32(float* __restrict__ p, int n) {
  int i = blockIdx.x * blockDim.x + threadIdx.x;
  if (i < n) p[i] = 0.0f;
}

// --------------------------------------------- sigmoid(w) + degree counting
__global__ void gnn_edge_prep(const float* __restrict__ ew,
                              const long long* __restrict__ ei,  // [2,E] int64
                              float* __restrict__ w,
                              float* __restrict__ cnt_dst,
                              float* __restrict__ cnt_src, int E) {
  int e = blockIdx.x * blockDim.x + threadIdx.x;
  if (e >= E) return;
  float v = ew[e];
  w[e] = 1.0f / (1.0f + __expf(-v));
  int src = (int)ei[e];
  int dst = (int)ei[(long)E + e];
  unsafeAtomicAdd(&cnt_dst[dst], 1.0f);   // global_atomic_add_f32
  unsafeAtomicAdd(&cnt_src[src], 1.0f);
}

// --------------------------------------------------- weighted edge scatter
// One wave per edge: lane holds 4 consecutive f32 of the 128-wide feature row.
__global__ void gnn_edge_scatter(const float* __restrict__ x,
                                 const long long* __restrict__ gat,   // gather idx
                                 const long long* __restrict__ sct,   // scatter idx
                                 const float* __restrict__ w,
                                 float* __restrict__ agg, int E) {
  int gid = blockIdx.x * blockDim.x + threadIdx.x;
  int e = gid >> 5;
  if (e >= E) return;
  int lane = gid & 31;
  int g = (int)gat[e];
  int s = (int)sct[e];
  float we = w[e];
  const float4 v = ((const float4*)(x + (long)g * DIM))[lane];
  float* ar = agg + (long)s * DIM + lane * 4;
  unsafeAtomicAdd(ar + 0, v.x * we);
  unsafeAtomicAdd(ar + 1, v.y * we);
  unsafeAtomicAdd(ar + 2, v.z * we);
  unsafeAtomicAdd(ar + 3, v.w * we);
}

// ------------------------------------------------ fused mean+GEMM+bias node op
// out[i,:] = act( (agg[i,:]/max(cnt[i],1)) @ Wl^T + bl + x[i,:] @ Wr^T )
// block = 256 thr (8 waves); block owns 16 node rows; wave owns 16 output cols.
__global__ void __launch_bounds__(256)
gnn_sage_linear(const float* __restrict__ agg, const float* __restrict__ cnt,
                const float* __restrict__ x,
                const float* __restrict__ Wl, const float* __restrict__ bl,
                const float* __restrict__ Wr,
                float* __restrict__ out, int n, int do_relu) {
  __shared__ float shMean[16 * LPITCH];
  __shared__ float shX[16 * LPITCH];

  const int tid = threadIdx.x;
  const int rowBase = blockIdx.x * 16;

  // Stage 16 rows of mean (agg * 1/max(cnt,1)) and x into LDS.
  for (int i = tid; i < 16 * DIM; i += 256) {
    int r = i >> 7, c = i & 127;
    int gr = rowBase + r;
    float m = 0.0f, xv = 0.0f;
    if (gr < n) {
      float inv = 1.0f / fmaxf(cnt[gr], 1.0f);
      m  = agg[(long)gr * DIM + c] * inv;
      xv = x[(long)gr * DIM + c];
    }
    shMean[r * LPITCH + c] = m;
    shX[r * LPITCH + c]   = xv;
  }
  __syncthreads();

  const int wave = tid >> 5;
  const int lane = tid & 31;
  const int nIdx = lane & 15;      // A row (M) and B/D column (N)
  const int half = lane >> 4;      // K sub-block for A/B, row offset for D
  const int colBase = wave * 16;
  const int kk = half * 2;

  // Accumulator seeded with bias (same bias across all 16 rows of the tile).
  v8f c;
  {
    float b = bl[colBase + nIdx];
#pragma unroll
    for (int r = 0; r < 8; ++r) c[r] = b;
  }

  const float* aMean = shMean + nIdx * LPITCH + kk;
  const float* aX    = shX    + nIdx * LPITCH + kk;
  const float* wlRow = Wl + (long)(colBase + nIdx) * DIM + kk;  // B[k][n] = Wl[n][k]
  const float* wrRow = Wr + (long)(colBase + nIdx) * DIM + kk;

#pragma unroll 4
  for (int k = 0; k < DIM; k += 4) {
    v2f a, b;
    // mean @ Wl^T
    a[0] = aMean[k];     a[1] = aMean[k + 1];
    b[0] = wlRow[k];     b[1] = wlRow[k + 1];
    c = __builtin_amdgcn_wmma_f32_16x16x4_f32(false, a, false, b,
                                              (short)0, c, false, false);
    // x @ Wr^T
    a[0] = aX[k];        a[1] = aX[k + 1];
    b[0] = wrRow[k];     b[1] = wrRow[k + 1];
    c = __builtin_amdgcn_wmma_f32_16x16x4_f32(false, a, false, b,
                                              (short)0, c, false, false);
  }

  // D layout: VGPR r -> row (r + 8*half), column nIdx.
#pragma unroll
  for (int r = 0; r < 8; ++r) {
    int gr = rowBase + r + 8 * half;
    if (gr < n) {
      float v = c[r];
      if (do_relu) v = fmaxf(v, 0.0f);
      out[(long)gr * DIM + colBase + nIdx] = v;
    }
  }
}

// ---------------------------------------------------------------------------
extern "C" void kernel_launch(void* const* d_in, const int* in_sizes, int n_in,
                              void* d_out, int out_size, void* d_ws, size_t ws_size,
                              hipStream_t stream) {
  const float*     s_feat = (const float*)d_in[0];
  const float*     t_feat = (const float*)d_in[1];
  const long long* ei     = (const long long*)d_in[2];   // int64 [2,E]
  const float*     ew     = (const float*)d_in[3];
  const float* s0_Wl = (const float*)d_in[4];
  const float* s0_bl = (const float*)d_in[5];
  const float* s0_Wr = (const float*)d_in[6];
  const float* t0_Wl = (const float*)d_in[7];
  const float* t0_bl = (const float*)d_in[8];
  const float* t0_Wr = (const float*)d_in[9];
  const float* s1_Wl = (const float*)d_in[10];
  const float* s1_bl = (const float*)d_in[11];
  const float* s1_Wr = (const float*)d_in[12];
  const float* t1_Wl = (const float*)d_in[13];
  const float* t1_bl = (const float*)d_in[14];
  const float* t1_Wr = (const float*)d_in[15];

  const int N = in_sizes[0] / DIM;     // 50000
  const int E = in_sizes[3];           // 800000
  const long long* src = ei;           // edge_index[0]
  const long long* dst = ei + E;       // edge_index[1]

  // ---- workspace carving (256B aligned) -----------------------------------
  size_t off = 0;
  auto carve = [&](size_t bytes) -> float* {
    off = (off + 255) & ~(size_t)255;
    float* p = (float*)((char*)d_ws + off);
    off += bytes;
    return p;
  };
  float* w       = carve((size_t)E * 4);          //  3.2 MB
  float* cnt_dst = carve((size_t)N * 4);          //  0.2 MB
  float* cnt_src = carve((size_t)N * 4);          //  0.2 MB
  float* agg     = carve((size_t)N * DIM * 4);    // 25.6 MB
  float* s1_ws   = carve((size_t)N * DIM * 4);    // 25.6 MB
  (void)ws_size;

  float* out_s = (float*)d_out;                   // s2
  float* out_t = (float*)d_out + (size_t)N * DIM; // t1 (temp) then t2

  const int ZB = 256;
  const int ND = N * DIM;
  dim3 zeroNDGrid((ND + ZB - 1) / ZB);
  dim3 edgeGrid((E + ZB - 1) / ZB);
  dim3 scatGrid(((size_t)E * 32 + ZB - 1) / ZB);
  dim3 linGrid((N + 15) / 16);

  // ---- edge preprocessing: sigmoid weights + degree counts ----------------
  gnn_zero_f32<<<dim3((2 * N + ZB - 1) / ZB), ZB, 0, stream>>>(cnt_dst, 2 * N); // cnt_src starts at next 256B boundary: adjacent only if N % 64 == 0
  gnn_edge_prep<<<edgeGrid, ZB, 0, stream>>>(ew, ei, w, cnt_dst, cnt_src, E);

  // ---- layer 0: s1 = relu(sage(t, src->dst)) -> ws ------------------------
  gnn_zero_f32<<<zeroNDGrid, ZB, 0, stream>>>(agg, ND);
  gnn_edge_scatter<<<scatGrid, ZB, 0, stream>>>(t_feat, src, dst, w, agg, E);
  gnn_sage_linear<<<linGrid, ZB, 0, stream>>>(agg, cnt_dst, t_feat,
                                              s0_Wl, s0_bl, s0_Wr, s1_ws, N, 1);

  // ---- layer 0: t1 = relu(sage(s, dst->src)) -> d_out upper (temp) --------
  gnn_zero_f32<<<zeroNDGrid, ZB, 0, stream>>>(agg, ND);
  gnn_edge_scatter<<<scatGrid, ZB, 0, stream>>>(s_feat, dst, src, w, agg, E);
  gnn_sage_linear<<<linGrid, ZB, 0, stream>>>(agg, cnt_src, s_feat,
                                              t0_Wl, t0_bl, t0_Wr, out_t, N, 1);

  // ---- layer 1: s2 = sage(t1, src->dst) -> d_out lower --------------------
  gnn_zero_f32<<<zeroNDGrid, ZB, 0, stream>>>(agg, ND);
  gnn_edge_scatter<<<scatGrid, ZB, 0, stream>>>(out_t, src, dst, w, agg, E);
  gnn_sage_linear<<<linGrid, ZB, 0, stream>>>(agg, cnt_dst, out_t,
                                              s1_Wl, s1_bl, s1_Wr, out_s, N, 0);

  // ---- layer 1: t2 = sage(s1, dst->src) -> d_out upper (t1 consumed) ------
  gnn_zero_f32<<<zeroNDGrid, ZB, 0, stream>>>(agg, ND);
  gnn_edge_scatter<<<scatGrid, ZB, 0, stream>>>(s1_ws, dst, src, w, agg, E);
  gnn_sage_linear<<<linGrid, ZB, 0, stream>>>(agg, cnt_src, s1_ws,
                                              t1_Wl, t1_bl, t1_Wr, out_t, N, 0);
}